// CadDeformLayer_63075889709151
// MI455X (gfx1250) — compile-verified
//
#include <hip/hip_runtime.h>

#define GRID_DIM 64

typedef __attribute__((ext_vector_type(2))) float v2f;
typedef __attribute__((ext_vector_type(8))) float v8f;

// ---------------- reductions ----------------
__device__ __forceinline__ float waveReduceSum(float v) {
#pragma unroll
    for (int off = 16; off > 0; off >>= 1)
        v += __shfl_down(v, off, 32);
    return v;
}

// blockDim.x == 256 (8 wave32)
__device__ __forceinline__ float blockReduceSum(float v) {
    __shared__ float smem[8];
    const int lane = threadIdx.x & 31;
    const int wave = threadIdx.x >> 5;
    v = waveReduceSum(v);
    if (lane == 0) smem[wave] = v;
    __syncthreads();
    float r = 0.0f;
    if (wave == 0) {
        r = (lane < 8) ? smem[lane] : 0.0f;
        r = waveReduceSum(r);
    }
    return r; // valid in thread 0
}

// ---------------- kernel A: trilinear distance-grid loss ----------------
__global__ void __launch_bounds__(256)
loss_grid_kernel(const float* __restrict__ V, const float* __restrict__ grid,
                 float* __restrict__ ws, int N) {
    int i = blockIdx.x * blockDim.x + threadIdx.x;
    const int stride = gridDim.x * blockDim.x;
    float acc = 0.0f;
    for (; i < N; i += stride) {
        // prefetch next streamed point (speculative; dropped if OOB)
        __builtin_prefetch(&V[3 * (i + stride)], 0, 0);
        const float px = V[3 * i + 0] * (float)(GRID_DIM - 1);
        const float py = V[3 * i + 1] * (float)(GRID_DIM - 1);
        const float pz = V[3 * i + 2] * (float)(GRID_DIM - 1);
        int x0 = (int)floorf(px); x0 = x0 < 0 ? 0 : (x0 > GRID_DIM - 2 ? GRID_DIM - 2 : x0);
        int y0 = (int)floorf(py); y0 = y0 < 0 ? 0 : (y0 > GRID_DIM - 2 ? GRID_DIM - 2 : y0);
        int z0 = (int)floorf(pz); z0 = z0 < 0 ? 0 : (z0 > GRID_DIM - 2 ? GRID_DIM - 2 : z0);
        const float fx = px - (float)x0;
        const float fy = py - (float)y0;
        const float fz = pz - (float)z0;
        // grid strides: x -> 4096, y -> 64, z -> 1 (1 MB table: L2-resident)
        const float* b = grid + (((x0 * GRID_DIM) + y0) * GRID_DIM + z0);
        const float g000 = b[0],                       g001 = b[1];
        const float g010 = b[GRID_DIM],                g011 = b[GRID_DIM + 1];
        const float g100 = b[GRID_DIM * GRID_DIM],     g101 = b[GRID_DIM * GRID_DIM + 1];
        const float g110 = b[GRID_DIM * GRID_DIM + GRID_DIM],
                    g111 = b[GRID_DIM * GRID_DIM + GRID_DIM + 1];
        const float c00 = g000 * (1.0f - fx) + g100 * fx;
        const float c01 = g001 * (1.0f - fx) + g101 * fx;
        const float c10 = g010 * (1.0f - fx) + g110 * fx;
        const float c11 = g011 * (1.0f - fx) + g111 * fx;
        const float c0 = c00 * (1.0f - fy) + c10 * fy;
        const float c1 = c01 * (1.0f - fy) + c11 * fy;
        const float d = c0 * (1.0f - fz) + c1 * fz;
        acc += 0.5f * d * d;
    }
    const float s = blockReduceSum(acc);
    if (threadIdx.x == 0) ws[blockIdx.x] = s;
}

// ---------------- kernel B: edge rest-length loss ----------------
__global__ void __launch_bounds__(256)
loss_edge_kernel(const float* __restrict__ V, const int* __restrict__ E,
                 const float* __restrict__ rest, float* __restrict__ ws, int Ecnt) {
    int e = blockIdx.x * blockDim.x + threadIdx.x;
    const int stride = gridDim.x * blockDim.x;
    float acc = 0.0f;
    for (; e < Ecnt; e += stride) {
        // E and rest are touched exactly once: non-temporal so they don't
        // evict the src_V gather set from L2.
        const int i0 = __builtin_nontemporal_load(&E[2 * e + 0]);
        const int i1 = __builtin_nontemporal_load(&E[2 * e + 1]);
        const float r = __builtin_nontemporal_load(&rest[e]);
        const float dx = V[3 * i0 + 0] - V[3 * i1 + 0];
        const float dy = V[3 * i0 + 1] - V[3 * i1 + 1];
        const float dz = V[3 * i0 + 2] - V[3 * i1 + 2];
        const float len = sqrtf(dx * dx + dy * dy + dz * dz + 1e-12f);
        const float t = len - r;
        acc += 0.5f * t * t;
    }
    const float s = blockReduceSum(acc);
    if (threadIdx.x == 0) ws[blockIdx.x] = s;
}

// ---------------- final deterministic reduce via V_WMMA_F32_16X16X4_F32 ----
// One wave32, EXEC all ones. A = ones(16x4) so D = colsum(B) + C; each WMMA
// folds 64 f32 partials into the f32 accumulator tile with zero precision
// loss. total must be a multiple of 64 and every slot is written by A/B.
__global__ void __launch_bounds__(32)
final_reduce_wmma(const float* __restrict__ ws, float* __restrict__ out, int total) {
    const int lane = threadIdx.x;         // 0..31
    const int n = lane & 15;              // column
    const int h = lane >> 4;              // half-wave selects K rows {0,1} vs {2,3}
    v2f a; a[0] = 1.0f; a[1] = 1.0f;      // all-ones A (16x4)
    v8f c = {};                           // f32 accumulator tile
    for (int base = 0; base < total; base += 64) {
        v2f b;
        b[0] = ws[base + (2 * h + 0) * 16 + n];
        b[1] = ws[base + (2 * h + 1) * 16 + n];
        c = __builtin_amdgcn_wmma_f32_16x16x4_f32(
                /*neg_a=*/false, a, /*neg_b=*/false, b,
                /*c_mod=*/(short)0, c, /*reuse_a=*/false, /*reuse_b=*/false);
    }
    // Row M=0 lives in VGPR0 of lanes 0..15; sum its 16 columns.
    float v = (lane < 16) ? c[0] : 0.0f;
    v = waveReduceSum(v);
    if (lane == 0) out[0] = v;
}

// ---------------- launch ----------------
extern "C" void kernel_launch(void* const* d_in, const int* in_sizes, int n_in,
                              void* d_out, int out_size, void* d_ws, size_t ws_size,
                              hipStream_t stream) {
    const float* V    = (const float*)d_in[0];   // [N,3] f32
    // d_in[1] = src_F : unused by the reference
    const int*   E    = (const int*)d_in[2];     // [E,2] int
    const float* grid = (const float*)d_in[3];   // [64,64,64] f32
    const float* rest = (const float*)d_in[4];   // [E] f32
    float* out = (float*)d_out;
    float* ws  = (float*)d_ws;

    const int N    = in_sizes[0] / 3;
    const int Ecnt = in_sizes[2] / 2;

    constexpr int NA = 2048;                 // partials from grid-loss pass
    constexpr int NB = 4096;                 // partials from edge-loss pass
    // NA+NB = 6144 = 96 * 64  -> exact WMMA chunks, 24 KB of d_ws

    loss_grid_kernel<<<NA, 256, 0, stream>>>(V, grid, ws, N);
    loss_edge_kernel<<<NB, 256, 0, stream>>>(V, E, rest, ws + NA, Ecnt);
    final_reduce_wmma<<<1, 32, 0, stream>>>(ws, out, NA + NB);
}